// TranscoderSet_71219147702401
// MI455X (gfx1250) — compile-verified
//
#include <hip/hip_runtime.h>
#include <hip/hip_bf16.h>

// select_decoder_vectors on MI455X (gfx1250, wave32):
//   out[i, :] = values[i] * W_D[layer_idx[i], feat_idx[i], :]
// W_D (12, 16384, 768) fp32, nnz = 131072, out (nnz, 768) fp32.
//
// Bandwidth-bound gather+scale (~805 MB total traffic, ~35 us floor at
// 23.3 TB/s). CDNA5-idiomatic pipeline:
//   * one wave32 per gathered row (row base address is wave-uniform)
//   * rows staged global->LDS with GLOBAL_LOAD_ASYNC_TO_LDS_B128 (ASYNCcnt),
//     6 x 512B per row, double-buffered in LDS (2 x 3072B per wave)
//   * explicit s_wait_asynccnt pipeline: issue row k+1, wait <=6 so row k is
//     complete (async loads retire in order), consume row k from LDS
//   * ds_load_b128 readback, scale, nontemporal global_store_b128 (write-once
//     output stays out of the 192MB L2; gathered W_D rows keep their reuse)

typedef float v4f __attribute__((ext_vector_type(4)));

#define D_SAE     16384
#define D_MODEL   768
#define WAVE      32
#define BLOCK     256                 // 8 waves per block
#define RPW       4                   // rows (nnz entries) per wave
#define ROW_BYTES (D_MODEL * 4)       // 3072 B per decoder row

__global__ __launch_bounds__(BLOCK) void select_decoder_vectors_kernel(
    const float* __restrict__ W_D,
    const float* __restrict__ values,
    const int*   __restrict__ layer_idx,
    const int*   __restrict__ feat_idx,
    float*       __restrict__ out,
    int nnz)
{
    // Per-wave double buffer: 8 waves * 2 * 3072B = 48 KB (of 320 KB WGP LDS)
    __shared__ __align__(16) unsigned char lds_buf[BLOCK / WAVE][2][ROW_BYTES];

    const int lane  = threadIdx.x & (WAVE - 1);
    const int wslot = threadIdx.x >> 5;
    const int wid   = blockIdx.x * (BLOCK / WAVE) + wslot;
    const int row0  = wid * RPW;
    if (row0 >= nnz) return;

    const int nrows = (nnz - row0 < RPW) ? (nnz - row0) : RPW;

    // Gather per-row metadata and source pointers up front.
    const char* src[RPW];
    float       scale[RPW];
#pragma unroll
    for (int k = 0; k < RPW; ++k) {
        if (k < nrows) {
            const int r = row0 + k;
            src[k] = reinterpret_cast<const char*>(
                W_D + ((size_t)layer_idx[r] * (size_t)D_SAE +
                       (size_t)feat_idx[r]) * (size_t)D_MODEL);
            scale[k] = values[r];
        }
    }

    // Raw LDS byte addresses: low 32 bits of the flat pointer into the LDS
    // aperture are the LDS offset.
    const unsigned lbase0 = (unsigned)(uintptr_t)(void*)&lds_buf[wslot][0][0];
    const unsigned lbase1 = (unsigned)(uintptr_t)(void*)&lds_buf[wslot][1][0];
    const unsigned lane16 = (unsigned)lane * 16u;

    // Issue one row's 6 async 512B transfers. INST_OFFSET is added to BOTH
    // the LDS and global addresses, so one LDS-addr VGPR + one 64-bit global
    // address VGPR pair covers the whole 3072B row.
    auto issue_row = [&](int k, unsigned lbase) {
        const char*    g = src[k] + lane16;   // per-lane 16B slice
        const unsigned l = lbase + lane16;
        asm volatile(
            "global_load_async_to_lds_b128 %0, %1, off offset:0\n\t"
            "global_load_async_to_lds_b128 %0, %1, off offset:512\n\t"
            "global_load_async_to_lds_b128 %0, %1, off offset:1024\n\t"
            "global_load_async_to_lds_b128 %0, %1, off offset:1536\n\t"
            "global_load_async_to_lds_b128 %0, %1, off offset:2048\n\t"
            "global_load_async_to_lds_b128 %0, %1, off offset:2560"
            :
            : "v"(l), "v"(g)
            : "memory");
    };

    // Prime the pipeline: row 0 -> buffer 0.
    issue_row(0, lbase0);

#pragma unroll
    for (int k = 0; k < RPW; ++k) {
        if (k < nrows) {
            if (k + 1 < nrows) {
                // Row k+1 -> buffer (k+1)&1, then wait until only those 6
                // async ops remain outstanding => row k fully landed in LDS.
                issue_row(k + 1, (k & 1) ? lbase0 : lbase1);
                asm volatile("s_wait_asynccnt 0x6" ::: "memory");
            } else {
                asm volatile("s_wait_asynccnt 0x0" ::: "memory");
            }

            const v4f* lb = reinterpret_cast<const v4f*>(
                &lds_buf[wslot][k & 1][0]);
            v4f* dst = reinterpret_cast<v4f*>(
                out + (size_t)(row0 + k) * (size_t)D_MODEL);
            const float vk = scale[k];

            // 6 x ds_load_b128 readback, scale, nontemporal b128 store.
#pragma unroll
            for (int j = 0; j < D_MODEL / (4 * WAVE); ++j) {   // 6 iters
                v4f x = lb[j * WAVE + lane];
                x *= vk;
                __builtin_nontemporal_store(x, dst + j * WAVE + lane);
            }
            // All ds_loads of this iteration are consumed above (compiler
            // inserts s_wait_dscnt before use), so the next iteration's async
            // overwrite of this buffer cannot race the reads.
        }
    }
}

extern "C" void kernel_launch(void* const* d_in, const int* in_sizes, int n_in,
                              void* d_out, int out_size, void* d_ws, size_t ws_size,
                              hipStream_t stream) {
    // setup_inputs() order: W_D, values, layer_idx, pos_idx, feat_idx
    const float* W_D       = (const float*)d_in[0];
    const float* values    = (const float*)d_in[1];
    const int*   layer_idx = (const int*)d_in[2];
    // d_in[3] = pos_idx: not used by the reference math
    const int*   feat_idx  = (const int*)d_in[4];
    float*       out       = (float*)d_out;

    const int nnz = in_sizes[1];                       // 131072
    const int rows_per_block = (BLOCK / WAVE) * RPW;   // 32 rows / block
    const int grid = (nnz + rows_per_block - 1) / rows_per_block;

    select_decoder_vectors_kernel<<<grid, BLOCK, 0, stream>>>(
        W_D, values, layer_idx, feat_idx, out, nnz);
}